// Pointnet2_76562087019210
// MI455X (gfx1250) — compile-verified
//
#include <hip/hip_runtime.h>
#include <stdint.h>

typedef float v2f __attribute__((ext_vector_type(2)));
typedef float v8f __attribute__((ext_vector_type(8)));

static __device__ __forceinline__ v8f zero8() {
  v8f z;
  for (int i = 0; i < 8; ++i) z[i] = 0.0f;
  return z;
}

// D = A(16x4 f32) * B(4x16 f32) + C(16x16 f32), full-wave op.
static __device__ __forceinline__ v8f wmma4(v2f a, v2f b, v8f c) {
  return __builtin_amdgcn_wmma_f32_16x16x4_f32(false, a, false, b, (short)0, c, false, false);
}

// ---------------------------------------------------------------------------
// Weight folding: Wt[k*cout+n] = W[n*cin+k] * gamma[n]/sqrt(var[n]+eps),
// rows k in [cin,cinPad) zero.  bo[n] = (b[n]-mean[n])*scale + beta[n].
// ---------------------------------------------------------------------------
__global__ __launch_bounds__(256) void fold_bn(
    const float* __restrict__ W, const float* __restrict__ b,
    const float* __restrict__ beta, const float* __restrict__ gamma,
    const float* __restrict__ mean, const float* __restrict__ var,
    int cin, int cinPad, int cout, float* __restrict__ Wt, float* __restrict__ bo)
{
  int i = blockIdx.x * 256 + threadIdx.x;
  if (i < cout) {
    float sc = gamma[i] * rsqrtf(var[i] + 1e-5f);
    bo[i] = (b[i] - mean[i]) * sc + beta[i];
  }
  if (i < cinPad * cout) {
    int k = i / cout, n = i - k * cout;
    float sc = gamma[n] * rsqrtf(var[n] + 1e-5f);
    Wt[i] = (k < cin) ? W[(size_t)n * cin + k] * sc : 0.0f;
  }
}

__global__ __launch_bounds__(256) void fold_plain(
    const float* __restrict__ W, const float* __restrict__ b,
    int cin, int cout, float* __restrict__ Wt, float* __restrict__ bo)
{
  int i = blockIdx.x * 256 + threadIdx.x;
  if (i < cout) bo[i] = b[i];
  if (i < cin * cout) {
    int k = i / cout, n = i - k * cout;
    Wt[i] = W[(size_t)n * cin + k];
  }
}

// ---------------------------------------------------------------------------
// Farthest point sampling: one block per batch; first-index tie-break argmax.
// ---------------------------------------------------------------------------
__global__ __launch_bounds__(256) void fps_kernel(
    const float* __restrict__ xyz, int stride, int N, int npoint, int* __restrict__ outIdx)
{
  __shared__ float dist[4096];
  __shared__ float bv[256];
  __shared__ int   bi[256];
  int b = blockIdx.x;
  int tid = threadIdx.x;
  const float* base = xyz + (size_t)b * N * stride;
  for (int i = tid; i < N; i += 256) dist[i] = 1e10f;
  __syncthreads();
  int far = 0;
  for (int it = 0; it < npoint; ++it) {
    if (tid == 0) outIdx[b * npoint + it] = far;
    float cx = base[(size_t)far * stride];
    float cy = base[(size_t)far * stride + 1];
    float cz = base[(size_t)far * stride + 2];
    float best = -1.0f; int bidx = 0;
    for (int i = tid; i < N; i += 256) {
      const float* p = base + (size_t)i * stride;
      float dx = p[0] - cx, dy = p[1] - cy, dz = p[2] - cz;
      float d = dx * dx + dy * dy + dz * dz;
      float nd = fminf(dist[i], d);
      dist[i] = nd;
      if (nd > best || (nd == best && i < bidx)) { best = nd; bidx = i; }
    }
    bv[tid] = best; bi[tid] = bidx;
    __syncthreads();
    for (int off = 128; off > 0; off >>= 1) {
      if (tid < off) {
        float ov = bv[tid + off]; int oi = bi[tid + off];
        if (ov > bv[tid] || (ov == bv[tid] && oi < bi[tid])) { bv[tid] = ov; bi[tid] = oi; }
      }
      __syncthreads();
    }
    far = bi[0];
    __syncthreads();
  }
}

// ---------------------------------------------------------------------------
// One MLP layer on a 32-row activation tile in LDS, weights [K][C] in global.
// 8 waves split the 2 x (C/16) WMMA C-tiles. bias + relu fused on store.
// ---------------------------------------------------------------------------
static __device__ __forceinline__ void mlp_layer(
    const float* __restrict__ actIn, int sIn, int K,
    const float* __restrict__ Wt, const float* __restrict__ bias, int C,
    float* __restrict__ actOut, int sOut, int wave, int lane)
{
  int mrow = lane & 15, h = lane >> 4;
  int total = 2 * (C >> 4);
  for (int t = wave; t < total; t += 8) {
    int mt = t & 1, nt = t >> 1;
    int col = nt * 16 + mrow;
    v8f acc = zero8();
    for (int kb = 0; kb < K; kb += 4) {
      int kk = kb + 2 * h;
      v2f a, w;
      const float* ar = actIn + (mt * 16 + mrow) * sIn + kk;
      a.x = ar[0]; a.y = ar[1];
      const float* wr = Wt + (size_t)kk * C + col;
      w.x = wr[0]; w.y = wr[C];
      acc = wmma4(a, w, acc);
    }
    float bvv = bias[col];
    for (int j = 0; j < 8; ++j) {
      float v = fmaxf(acc[j] + bvv, 0.0f);
      actOut[(mt * 16 + j + 8 * h) * sOut + col] = v;
    }
  }
}

// ---------------------------------------------------------------------------
// Fused set-abstraction: one block (256 thr) per sampled center.
//   wave0: ball query (ordered, fill-with-first)  -> sel[]
//   gather (centered xyz ++ point feats)          -> LDS
//   3 x (WMMA GEMM + BN(folded) + relu), layer3 fused with max over nsample.
// NSAMP processed in 32-row chunks to bound LDS.
// ---------------------------------------------------------------------------
template<int NSAMP, int CINP, int CP, int C1, int C2, int C3>
__global__ __launch_bounds__(256) void sa_kernel(
    const float* __restrict__ xyz, int N, int CX,
    const float* __restrict__ pts, int PSTR,
    const int* __restrict__ fpsIdx, int S,
    const float* __restrict__ Wt1, const float* __restrict__ b1,
    const float* __restrict__ Wt2, const float* __restrict__ b2,
    const float* __restrict__ Wt3, const float* __restrict__ b3,
    float radius2,
    float* __restrict__ outXyz, float* __restrict__ outPts)
{
  constexpr int S0  = CINP + 4;
  constexpr int S1  = C1 + 4;
  constexpr int S2  = C2 + 4;
  constexpr int CH  = NSAMP / 32;
  constexpr int NT3 = C3 / 16;     // multiple of 8 for our configs
  constexpr int NTW = NT3 / 8;
  constexpr int SZA = (32 * S0 > 32 * S2) ? 32 * S0 : 32 * S2;
  __shared__ int   sel[NSAMP];
  __shared__ float ctr[3];
  __shared__ float actA[SZA];      // gather buf, reused as layer-2 output
  __shared__ float actB[32 * S1];  // layer-1 output

  int bid = blockIdx.x;
  int b = bid / S, s = bid - b * S;
  int tid = threadIdx.x, wave = tid >> 5, lane = tid & 31;
  int mrow = lane & 15, h = lane >> 4;
  int fidx = fpsIdx[b * S + s];
  if (tid < 3) {
    float c = xyz[(size_t)(b * N + fidx) * CX + tid];
    ctr[tid] = c;
    outXyz[(size_t)(b * S + s) * 3 + tid] = c;
  }
  __syncthreads();

  if (tid < 32) {   // wave0: ordered ball query via ballot + prefix popcount
    float cx = ctr[0], cy = ctr[1], cz = ctr[2];
    int cnt = 0;
    int first = fidx;
    for (int c0 = 0; c0 < N && cnt < NSAMP; c0 += 32) {
      int i = c0 + lane;
      const float* p = xyz + (size_t)(b * N + i) * CX;
      float dx = p[0] - cx, dy = p[1] - cy, dz = p[2] - cz;
      unsigned mask = (unsigned)__ballot(dx * dx + dy * dy + dz * dz <= radius2);
      if (cnt == 0 && mask) first = c0 + __ffs(mask) - 1;
      int pos = cnt + __popc(mask & ((1u << lane) - 1u));
      if (((mask >> lane) & 1u) && pos < NSAMP) sel[pos] = i;
      cnt += __popc(mask);
    }
    if (cnt > NSAMP) cnt = NSAMP;
    for (int j = cnt + lane; j < NSAMP; j += 32) sel[j] = first;
  }
  __syncthreads();

  float rmax[NTW];
  for (int q = 0; q < NTW; ++q) rmax[q] = 0.0f;   // post-relu values are >= 0

  for (int chk = 0; chk < CH; ++chk) {
    // gather 32 group points into actA[m][k] (stride S0)
    for (int e = tid; e < 32 * CINP; e += 256) {
      int m = e / CINP, k = e - m * CINP;
      int idx = sel[chk * 32 + m];
      float v = 0.0f;
      if (k < 3)           v = xyz[(size_t)(b * N + idx) * CX + k] - ctr[k];
      else if (k < 3 + CP) v = pts[(size_t)(b * N + idx) * PSTR + (k - 3)];
      actA[m * S0 + k] = v;
    }
    __syncthreads();
    mlp_layer(actA, S0, CINP, Wt1, b1, C1, actB, S1, wave, lane);
    __syncthreads();
    mlp_layer(actB, S1, C1, Wt2, b2, C2, actA, S2, wave, lane);
    __syncthreads();
    // layer 3 + running column max: each wave owns full column strips
    for (int q = 0; q < NTW; ++q) {
      int nt = wave + q * 8;
      int col = nt * 16 + mrow;
      float bvv = b3[col];
      float rm = rmax[q];
      for (int mt = 0; mt < 2; ++mt) {
        v8f acc = zero8();
        for (int kb = 0; kb < C2; kb += 4) {
          int kk = kb + 2 * h;
          v2f a, w;
          const float* ar = actA + (mt * 16 + mrow) * S2 + kk;
          a.x = ar[0]; a.y = ar[1];
          const float* wr = Wt3 + (size_t)kk * C3 + col;
          w.x = wr[0]; w.y = wr[C3];
          acc = wmma4(a, w, acc);
        }
        for (int j = 0; j < 8; ++j) rm = fmaxf(rm, fmaxf(acc[j] + bvv, 0.0f));
      }
      rmax[q] = rm;
    }
    __syncthreads();
  }
  for (int q = 0; q < NTW; ++q) {
    int nt = wave + q * 8;
    float rm = rmax[q];
    rm = fmaxf(rm, __shfl_xor(rm, 16, 32));
    if (lane < 16) outPts[(size_t)(b * S + s) * C3 + nt * 16 + lane] = rm;
  }
}

// ---------------------------------------------------------------------------
// Generic WMMA GEMM: Y[M,N] = act(X[M,K] * Wt[K,N] + bias). 64x64 block tile,
// 32-wide K chunks staged in LDS, 8 waves x 2 C-tiles.
// ---------------------------------------------------------------------------
template<bool RELU>
static __device__ __forceinline__ void store_tile(
    const v8f& acc, float* __restrict__ Y, const float* __restrict__ bias,
    int mbase, int nbase, int M, int N, int mrow, int h)
{
  int col = nbase + mrow;
  if (col >= N) return;
  float bvv = bias[col];
  for (int j = 0; j < 8; ++j) {
    int row = mbase + j + 8 * h;
    if (row < M) {
      float v = acc[j] + bvv;
      if (RELU) v = fmaxf(v, 0.0f);
      Y[(size_t)row * N + col] = v;
    }
  }
}

template<bool RELU>
__global__ __launch_bounds__(256) void gemm_kernel(
    const float* __restrict__ X, const float* __restrict__ Wt,
    const float* __restrict__ bias, float* __restrict__ Y,
    int M, int N, int K)
{
  constexpr int AST = 36, BST = 68;
  __shared__ float As[64 * AST];
  __shared__ float Bs[32 * BST];
  int n0 = blockIdx.x * 64, m0 = blockIdx.y * 64;
  int tid = threadIdx.x, wave = tid >> 5, lane = tid & 31;
  int mrow = lane & 15, h = lane >> 4;
  int mtA = wave & 3,       ntA = wave >> 2;
  int mtB = (wave + 8) & 3, ntB = (wave + 8) >> 2;
  v8f acc0 = zero8(), acc1 = zero8();
  for (int kc = 0; kc < K; kc += 32) {
    for (int e = tid; e < 64 * 32; e += 256) {
      int m = e >> 5, k = e & 31;
      int gm = m0 + m, gk = kc + k;
      As[m * AST + k] = (gm < M && gk < K) ? X[(size_t)gm * K + gk] : 0.0f;
    }
    for (int e = tid; e < 32 * 64; e += 256) {
      int k = e >> 6, n = e & 63;
      int gk = kc + k, gn = n0 + n;
      Bs[k * BST + n] = (gk < K && gn < N) ? Wt[(size_t)gk * N + gn] : 0.0f;
    }
    __syncthreads();
    for (int ks = 0; ks < 32; ks += 4) {
      int kk = ks + 2 * h;
      v2f a, w;
      a.x = As[(mtA * 16 + mrow) * AST + kk]; a.y = As[(mtA * 16 + mrow) * AST + kk + 1];
      w.x = Bs[kk * BST + ntA * 16 + mrow];   w.y = Bs[(kk + 1) * BST + ntA * 16 + mrow];
      acc0 = wmma4(a, w, acc0);
      a.x = As[(mtB * 16 + mrow) * AST + kk]; a.y = As[(mtB * 16 + mrow) * AST + kk + 1];
      w.x = Bs[kk * BST + ntB * 16 + mrow];   w.y = Bs[(kk + 1) * BST + ntB * 16 + mrow];
      acc1 = wmma4(a, w, acc1);
    }
    __syncthreads();
  }
  store_tile<RELU>(acc0, Y, bias, m0 + mtA * 16, n0 + ntA * 16, M, N, mrow, h);
  store_tile<RELU>(acc1, Y, bias, m0 + mtB * 16, n0 + ntB * 16, M, N, mrow, h);
}

// SA3 last layer: GEMM (128 rows = one batch) + relu + max over the 128 rows.
// grid = (N/64, batch). K, N multiples of 32/64.
__global__ __launch_bounds__(256) void gemm_max_kernel(
    const float* __restrict__ X, const float* __restrict__ Wt,
    const float* __restrict__ bias, float* __restrict__ Y, int K, int N)
{
  constexpr int AST = 36, BST = 68;
  __shared__ float As[128 * AST];
  __shared__ float Bs[32 * BST];
  __shared__ float pm[128];
  int batch = blockIdx.y, n0 = blockIdx.x * 64;
  int tid = threadIdx.x, wave = tid >> 5, lane = tid & 31;
  int mrow = lane & 15, h = lane >> 4;
  int nt = wave & 3, mtb = (wave >> 2) * 4;
  v8f acc[4];
  for (int q = 0; q < 4; ++q) acc[q] = zero8();
  for (int kc = 0; kc < K; kc += 32) {
    for (int e = tid; e < 128 * 32; e += 256) {
      int m = e >> 5, k = e & 31;
      As[m * AST + k] = X[(size_t)(batch * 128 + m) * K + kc + k];
    }
    for (int e = tid; e < 32 * 64; e += 256) {
      int k = e >> 6, n = e & 63;
      Bs[k * BST + n] = Wt[(size_t)(kc + k) * N + n0 + n];
    }
    __syncthreads();
    for (int ks = 0; ks < 32; ks += 4) {
      int kk = ks + 2 * h;
      v2f w;
      w.x = Bs[kk * BST + nt * 16 + mrow];
      w.y = Bs[(kk + 1) * BST + nt * 16 + mrow];
      for (int q = 0; q < 4; ++q) {
        v2f a;
        int m = (mtb + q) * 16 + mrow;
        a.x = As[m * AST + kk]; a.y = As[m * AST + kk + 1];
        acc[q] = wmma4(a, w, acc[q]);
      }
    }
    __syncthreads();
  }
  float bvv = bias[n0 + nt * 16 + mrow];
  float rm = 0.0f;
  for (int q = 0; q < 4; ++q)
    for (int j = 0; j < 8; ++j) rm = fmaxf(rm, fmaxf(acc[q][j] + bvv, 0.0f));
  rm = fmaxf(rm, __shfl_xor(rm, 16, 32));
  if (lane < 16) pm[(wave >> 2) * 64 + nt * 16 + lane] = rm;
  __syncthreads();
  if (tid < 64) Y[(size_t)batch * N + n0 + tid] = fmaxf(pm[tid], pm[64 + tid]);
}

// feat3[r][c] = c<3 ? l2_xyz[r][c] : l2_pts[r][c-3]   (r = b*128+p, 259 cols)
__global__ __launch_bounds__(256) void gather3(
    const float* __restrict__ l2x, const float* __restrict__ l2p, float* __restrict__ feat)
{
  int i = blockIdx.x * 256 + threadIdx.x;
  if (i >= 32 * 128 * 259) return;
  int r = i / 259, c = i - r * 259;
  feat[i] = (c < 3) ? l2x[r * 3 + c] : l2p[(size_t)r * 256 + (c - 3)];
}

// ---------------------------------------------------------------------------
extern "C" void kernel_launch(void* const* d_in, const int* in_sizes, int n_in,
                              void* d_out, int out_size, void* d_ws, size_t ws_size,
                              hipStream_t stream)
{
  (void)in_sizes; (void)n_in; (void)out_size; (void)ws_size;
  // Flattened input order: xyz; then params with sorted dict keys:
  //   fc3{W,b}=1,2 ; head L0=3..8, L1=9..14 ; sa1 L0=15.., L1=21.., L2=27..
  //   sa2 L0=33.., L1=39.., L2=45.. ; sa3 L0=51.., L1=57.., L2=63..
  const float* xyz = (const float*)d_in[0];
  const float* fcW = (const float*)d_in[1];
  const float* fcB = (const float*)d_in[2];

  float* ws = (float*)d_ws;
  size_t off = 0;
  auto alloc = [&](size_t n) { float* p = ws + off; off += (n + 63) & ~(size_t)63; return p; };

  int*   fps1  = (int*)alloc(32 * 512);
  int*   fps2  = (int*)alloc(32 * 128);
  float* l1x   = alloc(32 * 512 * 3);
  float* l1p   = alloc((size_t)32 * 512 * 128);
  float* l2x   = alloc(32 * 128 * 3);
  float* l2p   = alloc((size_t)32 * 128 * 256);
  float* feat3 = alloc((size_t)4096 * 259);
  float* g3a   = alloc((size_t)4096 * 256);
  float* g3b   = alloc((size_t)4096 * 512);
  float* l3    = alloc(32 * 1024);
  float* h1    = alloc(32 * 512);
  float* h2    = alloc(32 * 256);

  float* w11 = alloc(8 * 64);      float* q11 = alloc(64);
  float* w12 = alloc(64 * 64);     float* q12 = alloc(64);
  float* w13 = alloc(64 * 128);    float* q13 = alloc(128);
  float* w21 = alloc(132 * 128);   float* q21 = alloc(128);
  float* w22 = alloc(128 * 128);   float* q22 = alloc(128);
  float* w23 = alloc(128 * 256);   float* q23 = alloc(256);
  float* w31 = alloc(259 * 256);   float* q31 = alloc(256);
  float* w32 = alloc(256 * 512);   float* q32 = alloc(512);
  float* w33 = alloc((size_t)512 * 1024); float* q33 = alloc(1024);
  float* wh1 = alloc((size_t)1024 * 512); float* qh1 = alloc(512);
  float* wh2 = alloc(512 * 256);   float* qh2 = alloc(256);
  float* wfc = alloc(256 * 40);    float* qfc = alloc(40);

  auto foldbn = [&](int base, int cin, int cinp, int cout, float* wt, float* bo) {
    int total = cinp * cout;
    fold_bn<<<(total + 255) / 256, 256, 0, stream>>>(
        (const float*)d_in[base],     (const float*)d_in[base + 1],
        (const float*)d_in[base + 2], (const float*)d_in[base + 3],
        (const float*)d_in[base + 4], (const float*)d_in[base + 5],
        cin, cinp, cout, wt, bo);
  };
  foldbn(15,   6,    8,   64, w11, q11);
  foldbn(21,  64,   64,   64, w12, q12);
  foldbn(27,  64,   64,  128, w13, q13);
  foldbn(33, 131,  132,  128, w21, q21);
  foldbn(39, 128,  128,  128, w22, q22);
  foldbn(45, 128,  128,  256, w23, q23);
  foldbn(51, 259,  259,  256, w31, q31);
  foldbn(57, 256,  256,  512, w32, q32);
  foldbn(63, 512,  512, 1024, w33, q33);
  foldbn(3, 1024, 1024,  512, wh1, qh1);
  foldbn(9,  512,  512,  256, wh2, qh2);
  fold_plain<<<(256 * 40 + 255) / 256, 256, 0, stream>>>(fcW, fcB, 256, 40, wfc, qfc);

  // SA1: FPS(512) on raw xyz (stride 6), then fused ball-query + MLP + max
  fps_kernel<<<32, 256, 0, stream>>>(xyz, 6, 4096, 512, fps1);
  sa_kernel<32, 8, 3, 64, 64, 128><<<32 * 512, 256, 0, stream>>>(
      xyz, 4096, 6, xyz + 3, 6, fps1, 512,
      w11, q11, w12, q12, w13, q13, 0.04f, l1x, l1p);

  // SA2: FPS(128) on l1_xyz (stride 3), nsample=64
  fps_kernel<<<32, 256, 0, stream>>>(l1x, 3, 512, 128, fps2);
  sa_kernel<64, 132, 128, 128, 128, 256><<<32 * 128, 256, 0, stream>>>(
      l1x, 512, 3, l1p, 128, fps2, 128,
      w21, q21, w22, q22, w23, q23, 0.16f, l2x, l2p);

  // SA3 (group_all): feat = [l2_xyz, l2_pts], 3 GEMMs, last fused with max.
  gather3<<<(32 * 128 * 259 + 255) / 256, 256, 0, stream>>>(l2x, l2p, feat3);
  gemm_kernel<true><<<dim3(4, 64), 256, 0, stream>>>(feat3, w31, q31, g3a, 4096, 256, 259);
  gemm_kernel<true><<<dim3(8, 64), 256, 0, stream>>>(g3a,   w32, q32, g3b, 4096, 512, 256);
  gemm_max_kernel<<<dim3(16, 32), 256, 0, stream>>>(g3b, w33, q33, l3, 512, 1024);

  // Head + final FC (writes d_out directly, fp32 [32,40])
  gemm_kernel<true ><<<dim3(8, 1), 256, 0, stream>>>(l3, wh1, qh1, h1, 32, 512, 1024);
  gemm_kernel<true ><<<dim3(4, 1), 256, 0, stream>>>(h1, wh2, qh2, h2, 32, 256, 512);
  gemm_kernel<false><<<dim3(1, 1), 256, 0, stream>>>(h2, wfc, qfc, (float*)d_out, 32, 40, 256);
}